// BackBoneGNN_47158740910488
// MI455X (gfx1250) — compile-verified
//
#include <hip/hip_runtime.h>
#include <math.h>

// ---------------------------------------------------------------------------
// BackBoneGNN for MI455X (gfx1250, wave32).
// Dense node transforms + JK projection use V_WMMA_F32_16X16X4_F32 (exact f32),
// fully unrolled with branchless (clamp+cndmask) padding so EXEC stays all-1s.
// Edge softmax/aggregation uses L2-resident native f32 atomics.
// ---------------------------------------------------------------------------

typedef float v2f __attribute__((ext_vector_type(2)));
typedef float v8f __attribute__((ext_vector_type(8)));

#define TPB 256
#define NUM_GRAPHS 1024

// ---------------- native float atomics ----------------
__device__ __forceinline__ void gAtomicAdd(float* p, float v) {
    __hip_atomic_fetch_add(p, v, __ATOMIC_RELAXED, __HIP_MEMORY_SCOPE_AGENT);
}
__device__ __forceinline__ void sAtomicAdd(float* p, float v) {
    __hip_atomic_fetch_add(p, v, __ATOMIC_RELAXED, __HIP_MEMORY_SCOPE_WORKGROUP);
}

// float atomic max via sign-split trick (init = -inf)
__device__ __forceinline__ void atomicMaxFloat(float* addr, float val) {
    if (val >= 0.0f) {
        atomicMax((int*)addr, __float_as_int(val));
    } else {
        atomicMin((unsigned int*)addr, __float_as_uint(val));
    }
}

// ---------------- WMMA: H[N,8] = X[N,LDX (K cols)] @ W[K,8] ----------------
// One wave per 16-node tile. M=16 nodes, N=16 cols (8 real + 8 zero pad),
// K consumed 4 at a time via v_wmma_f32_16x16x4_f32.
// A layout (16x4 f32): lanes 0-15 -> M=lane, K={kb,kb+1}; lanes 16-31 -> M=lane-16, K={kb+2,kb+3}
// D layout: lane l, vgpr j -> row = j + 8*(l>=16), col = l&15.
template <int K, int LDX>
__global__ void wmma_lin_kernel(const float* __restrict__ X,
                                const float* __restrict__ W,   // [K,8] row-major
                                float* __restrict__ Hout, int numTiles) {
    int wave = (blockIdx.x * blockDim.x + threadIdx.x) >> 5;   // uniform per wave
    if (wave >= numTiles) return;                              // wave-uniform branch
    const int lane = threadIdx.x & 31;
    const int half = lane >> 4;        // 0: K+0/1, 1: K+2/3
    const int l15  = lane & 15;
    const int col  = l15 & 7;          // clamped B column
    const bool bact = (l15 < 8);
    const float* xr = X + (size_t)(wave * 16 + l15) * LDX;

    v8f acc = {0.f, 0.f, 0.f, 0.f, 0.f, 0.f, 0.f, 0.f};
#pragma unroll
    for (int kb = 0; kb < K; kb += 4) {
        const int k0 = kb + (half ? 2 : 0);
        const int k1 = k0 + 1;
        v2f a, b;
        if (LDX == 8 && K == 8) {
            a = *(const v2f*)(xr + k0);                 // aligned global_load_b64
        } else {
            // branchless pad: clamp address, select value (no EXEC divergence)
            float ax = xr[(k0 < K) ? k0 : 0];
            float ay = xr[(k1 < K) ? k1 : 0];
            a.x = (k0 < K) ? ax : 0.0f;
            a.y = (k1 < K) ? ay : 0.0f;
        }
        float w0 = W[((k0 < K) ? k0 : 0) * 8 + col];
        float w1 = W[((k1 < K) ? k1 : 0) * 8 + col];
        b.x = (bact && k0 < K) ? w0 : 0.0f;
        b.y = (bact && k1 < K) ? w1 : 0.0f;
        acc = __builtin_amdgcn_wmma_f32_16x16x4_f32(false, a, false, b,
                                                    (short)0, acc, false, false);
    }
    if (bact) {
        int rowBase = wave * 16 + (half ? 8 : 0);
#pragma unroll
        for (int j = 0; j < 8; ++j)
            Hout[(size_t)(rowBase + j) * 8 + l15] = acc[j];
    }
}

// ---------------- per-node attention dot products ----------------
__global__ void dots_kernel(const float* __restrict__ H,
                            const float* __restrict__ as_, const float* __restrict__ ad_,
                            float* __restrict__ hs, float* __restrict__ hd, int N) {
    int n = blockIdx.x * blockDim.x + threadIdx.x;
    if (n >= N) return;
    float s = 0.f, d = 0.f;
#pragma unroll
    for (int c = 0; c < 8; ++c) {
        float v = H[(size_t)n * 8 + c];
        s += v * as_[c];
        d += v * ad_[c];
    }
    hs[n] = s;
    hd[n] = d;
}

// ---------------- per-layer init: acc=0, mmax=-inf, denom=0, stats=0 ----------------
__global__ void init_layer_kernel(float* __restrict__ accb, float* __restrict__ mmax,
                                  float* __restrict__ denom, float* __restrict__ stats, int N) {
    int i = blockIdx.x * blockDim.x + threadIdx.x;
    if (i < N * 8) accb[i] = 0.0f;
    if (i < N) { mmax[i] = -INFINITY; denom[i] = 0.0f; }
    if (i < 16) stats[i] = 0.0f;
}

__global__ void zero_pooled_kernel(float* __restrict__ pooled) {
    int i = blockIdx.x * blockDim.x + threadIdx.x;
    if (i < NUM_GRAPHS * 8) pooled[i] = 0.0f;
}

// ---------------- edge pass A: segment max of leaky_relu(hs[s]+hd[d]) ----------------
__global__ void edge_max_kernel(const int* __restrict__ ei, int E, int N,
                                const float* __restrict__ hs, const float* __restrict__ hd,
                                float* __restrict__ mmax) {
    int e = blockIdx.x * blockDim.x + threadIdx.x;
    if (e >= E + N) return;
    int s, d;
    if (e < E) { s = ei[e]; d = ei[E + e]; }
    else       { s = d = e - E; }                 // self loop
    float v = hs[s] + hd[d];
    v = (v > 0.f) ? v : 0.2f * v;                 // leaky_relu
    atomicMaxFloat(&mmax[d], v);
}

// ---------------- edge pass B: denom += ex ; acc[dst] += ex * h[src] ----------------
__global__ void edge_soft_kernel(const int* __restrict__ ei, int E, int N,
                                 const float* __restrict__ hs, const float* __restrict__ hd,
                                 const float* __restrict__ mmax,
                                 const float* __restrict__ Hlin,
                                 float* __restrict__ accb, float* __restrict__ denom) {
    int e = blockIdx.x * blockDim.x + threadIdx.x;
    if (e >= E + N) return;
    int s, d;
    if (e < E) { s = ei[e]; d = ei[E + e]; }
    else       { s = d = e - E; }
    float v = hs[s] + hd[d];
    v = (v > 0.f) ? v : 0.2f * v;
    float ex = expf(v - mmax[d]);
    gAtomicAdd(&denom[d], ex);
    const float4* hp = (const float4*)(Hlin + (size_t)s * 8);
    float4 h0 = hp[0], h1 = hp[1];
    float* ab = accb + (size_t)d * 8;
    gAtomicAdd(ab + 0, ex * h0.x);
    gAtomicAdd(ab + 1, ex * h0.y);
    gAtomicAdd(ab + 2, ex * h0.z);
    gAtomicAdd(ab + 3, ex * h0.w);
    gAtomicAdd(ab + 4, ex * h1.x);
    gAtomicAdd(ab + 5, ex * h1.y);
    gAtomicAdd(ab + 6, ex * h1.z);
    gAtomicAdd(ab + 7, ex * h1.w);
}

// ---------------- finalize GAT: out = acc/denom + bias (in place) ----------------
__global__ void finalize_kernel(float* __restrict__ accb, const float* __restrict__ denom,
                                const float* __restrict__ bias, int N) {
    int i = blockIdx.x * blockDim.x + threadIdx.x;
    if (i >= N * 8) return;
    int n = i >> 3, c = i & 7;
    accb[i] = accb[i] / (denom[n] + 1e-16f) + bias[c];
}

// ---------------- BN stats: per-channel sum / sumsq over N rows ----------------
__global__ void bn_stats_kernel(const float* __restrict__ H, float* __restrict__ stats, int N) {
    __shared__ float ssum[8], ssq[8];
    if (threadIdx.x < 8) { ssum[threadIdx.x] = 0.f; ssq[threadIdx.x] = 0.f; }
    __syncthreads();
    int n = blockIdx.x * blockDim.x + threadIdx.x;
    if (n < N) {
#pragma unroll
        for (int c = 0; c < 8; ++c) {
            float v = H[(size_t)n * 8 + c];
            sAtomicAdd(&ssum[c], v);
            sAtomicAdd(&ssq[c], v * v);
        }
    }
    __syncthreads();
    if (threadIdx.x < 8)  gAtomicAdd(&stats[threadIdx.x], ssum[threadIdx.x]);
    if (threadIdx.x >= 8 && threadIdx.x < 16)
        gAtomicAdd(&stats[threadIdx.x], ssq[threadIdx.x - 8]);
}

// ---------------- BN finalize: stats[16..23]=scale, stats[24..31]=shift ----------------
__global__ void bn_final_kernel(float* __restrict__ stats, const float* __restrict__ g,
                                const float* __restrict__ be, int N) {
    int c = threadIdx.x;
    if (c >= 8) return;
    float inv = 1.0f / (float)N;
    float mean = stats[c] * inv;
    float var  = stats[8 + c] * inv - mean * mean;   // biased variance (torch BN train)
    float rs   = rsqrtf(var + 1e-5f);
    float sc   = rs * g[c];
    stats[16 + c] = sc;
    stats[24 + c] = be[c] - mean * sc;
}

// ---------------- BN scale/shift + PReLU, in place ----------------
__global__ void act_kernel(float* __restrict__ H, const float* __restrict__ stats,
                           const float* __restrict__ aP, int N) {
    int i = blockIdx.x * blockDim.x + threadIdx.x;
    if (i >= N * 8) return;
    int c = i & 7;
    float v = H[i] * stats[16 + c] + stats[24 + c];
    float a = *aP;
    H[i] = (v > 0.f) ? v : a * v;
}

// ---------------- JK cat + projection (WMMA, K=24) + global_add_pool ----------------
__global__ void jk_pool_kernel(const float* __restrict__ h1, const float* __restrict__ h2,
                               const float* __restrict__ h3,
                               const float* __restrict__ Wjk,   // [24,8]
                               const float* __restrict__ bjk,
                               const int* __restrict__ batch,
                               float* __restrict__ pooled, int numTiles) {
    int wave = (blockIdx.x * blockDim.x + threadIdx.x) >> 5;
    if (wave >= numTiles) return;
    const int lane = threadIdx.x & 31;
    const int half = lane >> 4;
    const int l15  = lane & 15;
    const int col  = l15 & 7;
    const bool bact = (l15 < 8);
    const size_t mrow = (size_t)(wave * 16 + l15) * 8;

    const float* Hsrc[3] = {h1, h2, h3};
    v8f acc = {0.f, 0.f, 0.f, 0.f, 0.f, 0.f, 0.f, 0.f};
#pragma unroll
    for (int cc = 0; cc < 6; ++cc) {
        const float* Hb = Hsrc[cc >> 1];
        const int kloc = (cc & 1) * 4 + (half ? 2 : 0);   // local K in 8-wide block (even)
        const int kg   = cc * 4 + (half ? 2 : 0);         // global K in 0..23
        v2f a = *(const v2f*)(Hb + mrow + kloc);          // aligned global_load_b64
        v2f b;
        float w0 = Wjk[kg * 8 + col];
        float w1 = Wjk[(kg + 1) * 8 + col];
        b.x = bact ? w0 : 0.0f;
        b.y = bact ? w1 : 0.0f;
        acc = __builtin_amdgcn_wmma_f32_16x16x4_f32(false, a, false, b,
                                                    (short)0, acc, false, false);
    }
    if (bact) {
        float bj = bjk[l15];
        int rowBase = wave * 16 + (half ? 8 : 0);
#pragma unroll
        for (int j = 0; j < 8; ++j) {
            int row = rowBase + j;
            int g = batch[row];
            gAtomicAdd(&pooled[(size_t)g * 8 + l15], acc[j] + bj);
        }
    }
}

// ---------------- final MLP head on [1024,8] -> [1024,2], single block ----------------
__global__ void mlp_head_kernel(const float* __restrict__ pooled,
                                const float* __restrict__ Wm1, const float* __restrict__ bm1,
                                const float* __restrict__ gm, const float* __restrict__ bem,
                                const float* __restrict__ aP,
                                const float* __restrict__ Wm2, const float* __restrict__ bm2,
                                float* __restrict__ out) {
    __shared__ float ssum[8], ssq[8], ssc[8], ssh[8];
    int g = threadIdx.x;                       // 1024 threads = 1024 graphs
    if (g < 8) { ssum[g] = 0.f; ssq[g] = 0.f; }
    __syncthreads();

    float y[8];
#pragma unroll
    for (int c = 0; c < 8; ++c) {
        float accv = bm1[c];
#pragma unroll
        for (int k = 0; k < 8; ++k)
            accv += pooled[(size_t)g * 8 + k] * Wm1[k * 8 + c];
        y[c] = accv;
        sAtomicAdd(&ssum[c], accv);
        sAtomicAdd(&ssq[c], accv * accv);
    }
    __syncthreads();
    if (g < 8) {
        float mean = ssum[g] * (1.0f / NUM_GRAPHS);
        float var  = ssq[g] * (1.0f / NUM_GRAPHS) - mean * mean;
        float rs   = rsqrtf(var + 1e-5f);
        float sc   = rs * gm[g];
        ssc[g] = sc;
        ssh[g] = bem[g] - mean * sc;
    }
    __syncthreads();
    float a = *aP;
    float m1[8];
#pragma unroll
    for (int c = 0; c < 8; ++c) {
        float v = y[c] * ssc[c] + ssh[c];
        m1[c] = (v > 0.f) ? v : a * v;
    }
#pragma unroll
    for (int o = 0; o < 2; ++o) {
        float accv = bm2[o];
#pragma unroll
        for (int k = 0; k < 8; ++k)
            accv += m1[k] * Wm2[k * 2 + o];
        out[(size_t)g * 2 + o] = accv;
    }
}

// ---------------------------------------------------------------------------

extern "C" void kernel_launch(void* const* d_in, const int* in_sizes, int n_in,
                              void* d_out, int out_size, void* d_ws, size_t ws_size,
                              hipStream_t stream) {
    const float* x     = (const float*)d_in[0];
    const int*   ei    = (const int*)d_in[1];
    const int*   batch = (const int*)d_in[2];
    const float* W1 = (const float*)d_in[3];
    const float* as1 = (const float*)d_in[4];
    const float* ad1 = (const float*)d_in[5];
    const float* b1 = (const float*)d_in[6];
    const float* W2 = (const float*)d_in[7];
    const float* as2 = (const float*)d_in[8];
    const float* ad2 = (const float*)d_in[9];
    const float* b2 = (const float*)d_in[10];
    const float* W3 = (const float*)d_in[11];
    const float* as3 = (const float*)d_in[12];
    const float* ad3 = (const float*)d_in[13];
    const float* b3 = (const float*)d_in[14];
    const float* g1 = (const float*)d_in[15];
    const float* be1 = (const float*)d_in[16];
    const float* g2 = (const float*)d_in[17];
    const float* be2 = (const float*)d_in[18];
    const float* g3 = (const float*)d_in[19];
    const float* be3 = (const float*)d_in[20];
    const float* a_gnn = (const float*)d_in[21];
    const float* Wjk = (const float*)d_in[22];
    const float* bjk = (const float*)d_in[23];
    const float* Wm1 = (const float*)d_in[24];
    const float* bm1 = (const float*)d_in[25];
    const float* gm = (const float*)d_in[26];
    const float* bem = (const float*)d_in[27];
    const float* a_mlp = (const float*)d_in[28];
    const float* Wm2 = (const float*)d_in[29];
    const float* bm2 = (const float*)d_in[30];

    const int N = in_sizes[2];            // batch vector length = node count
    const int E = in_sizes[1] / 2;        // edge_index is [2,E]
    float* ws = (float*)d_ws;

    // workspace layout (floats)
    size_t off = 0;
    float* hlin  = ws + off; off += (size_t)N * 8;   // current layer pre-attention h
    float* h1    = ws + off; off += (size_t)N * 8;   // post-activation layer outputs
    float* h2    = ws + off; off += (size_t)N * 8;
    float* h3    = ws + off; off += (size_t)N * 8;
    float* hs    = ws + off; off += (size_t)N;
    float* hd    = ws + off; off += (size_t)N;
    float* mmax  = ws + off; off += (size_t)N;
    float* denom = ws + off; off += (size_t)N;
    float* stats = ws + off; off += 32;              // [sum8, sumsq8, scale8, shift8]
    float* pooled = ws + off; off += (size_t)NUM_GRAPHS * 8;

    const int numTiles = N / 16;                                 // N=500000 -> 31250 exact
    const int blkWmma  = (numTiles * 32 + TPB - 1) / TPB;
    const int blkN     = (N + TPB - 1) / TPB;
    const int blkN8    = (N * 8 + TPB - 1) / TPB;
    const int blkE     = (E + N + TPB - 1) / TPB;

    zero_pooled_kernel<<<(NUM_GRAPHS * 8 + TPB - 1) / TPB, TPB, 0, stream>>>(pooled);

    struct LayerP {
        const float* as_; const float* ad_; const float* bias;
        const float* g; const float* be; float* hout;
    };
    LayerP layers[3] = {
        { as1, ad1, b1, g1, be1, h1 },
        { as2, ad2, b2, g2, be2, h2 },
        { as3, ad3, b3, g3, be3, h3 },
    };

    for (int L = 0; L < 3; ++L) {
        LayerP& p = layers[L];
        // h = xin @ W   (WMMA f32 16x16x4, compile-time K/LDX)
        if (L == 0)      wmma_lin_kernel<7, 7><<<blkWmma, TPB, 0, stream>>>(x,  W1, hlin, numTiles);
        else if (L == 1) wmma_lin_kernel<8, 8><<<blkWmma, TPB, 0, stream>>>(h1, W2, hlin, numTiles);
        else             wmma_lin_kernel<8, 8><<<blkWmma, TPB, 0, stream>>>(h2, W3, hlin, numTiles);
        // hs = h @ a_src ; hd = h @ a_dst
        dots_kernel<<<blkN, TPB, 0, stream>>>(hlin, p.as_, p.ad_, hs, hd, N);
        // zero acc/denom/stats, mmax = -inf
        init_layer_kernel<<<blkN8, TPB, 0, stream>>>(p.hout, mmax, denom, stats, N);
        // segment max of attention logits
        edge_max_kernel<<<blkE, TPB, 0, stream>>>(ei, E, N, hs, hd, mmax);
        // softmax numerator/denominator accumulation
        edge_soft_kernel<<<blkE, TPB, 0, stream>>>(ei, E, N, hs, hd, mmax, hlin, p.hout, denom);
        // out = acc/denom + bias
        finalize_kernel<<<blkN8, TPB, 0, stream>>>(p.hout, denom, p.bias, N);
        // BatchNorm (train-mode, biased var) + PReLU, in place
        bn_stats_kernel<<<blkN, TPB, 0, stream>>>(p.hout, stats, N);
        bn_final_kernel<<<1, 8, 0, stream>>>(stats, p.g, p.be, N);
        act_kernel<<<blkN8, TPB, 0, stream>>>(p.hout, stats, a_gnn, N);
    }

    // JK 'cat' projection (WMMA, K=24) fused with global_add_pool
    jk_pool_kernel<<<blkWmma, TPB, 0, stream>>>(h1, h2, h3, Wjk, bjk, batch, pooled, numTiles);

    // final MLP head: BN + PReLU + linear to [G,2]
    mlp_head_kernel<<<1, NUM_GRAPHS, 0, stream>>>(pooled, Wm1, bm1, gm, bem, a_mlp,
                                                  Wm2, bm2, (float*)d_out);
}